// CAST_41274635715112
// MI455X (gfx1250) — compile-verified
//
#include <hip/hip_runtime.h>
#include <hip/hip_bf16.h>

typedef __attribute__((ext_vector_type(2))) float v2f;
typedef __attribute__((ext_vector_type(8))) float v8f;

#define Bc 64
#define Hc 50
#define Nc 50
#define Dc 400
#define Pc 100
#define Ac 200
#define Fc 500              // D + P
#define NEGV -1000000000.0f

// -------- 1) hist = concat(history_repr, pos_emb[pos]) : (B*H, 500) --------
__global__ void build_hist_kernel(const float* __restrict__ history,
                                  const int* __restrict__ pos,
                                  const float* __restrict__ pos_emb,
                                  float* __restrict__ hist) {
  int idx = blockIdx.x * blockDim.x + threadIdx.x;
  if (idx >= Bc * Hc * Fc) return;
  int f  = idx % Fc;
  int bh = idx / Fc;
  float v = (f < Dc) ? history[bh * Dc + f]
                     : pos_emb[pos[bh] * Pc + (f - Dc)];
  hist[idx] = v;
}

// -------- 2) cand = concat(candidate_repr, pos_emb[1]) : (B*N, 500) --------
// Written directly into the second output region.
__global__ void build_cand_kernel(const float* __restrict__ cand_repr,
                                  const float* __restrict__ pos_emb,
                                  float* __restrict__ cand) {
  int idx = blockIdx.x * blockDim.x + threadIdx.x;
  if (idx >= Bc * Nc * Fc) return;
  int f  = idx % Fc;
  int bn = idx / Fc;
  float v = (f < Dc) ? cand_repr[bn * Dc + f]
                     : pos_emb[Pc + (f - Dc)];        // pos_emb[1]
  cand[idx] = v;
}

// -------- 3) hc = cand*Wc^T, hh = hist*Wh^T via V_WMMA_F32_16X16X4_F32 -----
// One wave per 16x16 output tile. M=3200 (=200 tiles), N=200 (13 tiles, last
// one masked), K=500 (125 WMMA steps of K=4).
// A frag layout (16x4 f32): lanes 0-15 hold row M=lane, VGPR0/1 = K0/K1;
// lanes 16-31 hold K2/K3  -> one v2f load per lane at k = kb + 2*(lane>>4).
// B frag (4x16 f32) mirrors the K mapping with lanes holding N.
__global__ __launch_bounds__(32)
void gemm_wmma_kernel(const float* __restrict__ cand,  // (3200,500)
                      const float* __restrict__ hist,  // (3200,500)
                      const float* __restrict__ W1,    // (200,1000) row-major
                      float* __restrict__ hc,          // (3200,200)
                      float* __restrict__ hh) {        // (3200,200)
  const int lane  = threadIdx.x;
  const int lhalf = lane >> 4;          // 0 or 1
  const int lmod  = lane & 15;
  const int mtile = blockIdx.x;         // 0..199
  const int ntile = blockIdx.y;         // 0..12
  const int which = blockIdx.z;         // 0 -> hc, 1 -> hh

  const float* Arows = which ? hist : cand;
  const float* Wbase = W1 + (which ? Fc : 0);   // Wc = W1[:, :500], Wh = W1[:, 500:]
  float* Out = which ? hh : hc;

  const int mrow  = mtile * 16 + lmod;
  const int ncol  = ntile * 16 + lmod;
  const float nmask = (ncol < Ac) ? 1.0f : 0.0f;  // zero-pad A-dim to 208
  const int nclamp  = (ncol < Ac) ? ncol : (Ac - 1);

  const float* aptr = Arows + (size_t)mrow * Fc + 2 * lhalf;
  const float* bptr = Wbase + (size_t)nclamp * (2 * Fc) + 2 * lhalf;

  v8f c = {};
#pragma unroll 5
  for (int kb = 0; kb < Fc; kb += 4) {
    v2f a = *reinterpret_cast<const v2f*>(aptr);        // global_load_b64
    v2f b = *reinterpret_cast<const v2f*>(bptr);
    b.x *= nmask; b.y *= nmask;
    // (neg_a, A, neg_b, B, c_mod, C, reuse_a, reuse_b)
    c = __builtin_amdgcn_wmma_f32_16x16x4_f32(false, a, false, b,
                                              (short)0, c, false, false);
    aptr += 4;
    bptr += 4;
  }

  if (ncol < Ac) {
    const int mbase = mtile * 16 + lhalf * 8;
#pragma unroll
    for (int r = 0; r < 8; ++r)
      Out[(size_t)(mbase + r) * Ac + ncol] = c[r];
  }
}

// -------- 4) logits -> masked softmax -> user_repr = alpha * hist ----------
// One 64-thread block per (b,n).
__global__ __launch_bounds__(64)
void attn_kernel(const float* __restrict__ hc,    // (B*N,200)
                 const float* __restrict__ hh,    // (B*H,200)
                 const float* __restrict__ hist,  // (B*H,500)
                 const int* __restrict__ maskp,   // (B,H)
                 const float* __restrict__ b1,    // (200,)
                 const float* __restrict__ w2,    // (200,)
                 const float* __restrict__ b2,    // (1,)
                 float* __restrict__ out) {       // (B*N,500)
  __shared__ float s_hc[Ac];
  __shared__ float s_logit[Hc];
  __shared__ float s_alpha[Hc];
  __shared__ float s_red[2];

  const int tid = threadIdx.x;
  const int bn  = blockIdx.x;
  const int b   = bn / Nc;

  const float* hcrow = hc + (size_t)bn * Ac;
  for (int i = tid; i < Ac; i += 64) s_hc[i] = hcrow[i];
  __syncthreads();

  if (tid < Hc) {
    const float* hhrow = hh + (size_t)(b * Hc + tid) * Ac;
    float acc = 0.0f;
    for (int a = 0; a < Ac; ++a) {
      float v = s_hc[a] + hhrow[a] + b1[a];
      v = fmaxf(v, 0.0f);
      acc = fmaf(v, w2[a], acc);
    }
    acc += b2[0];
    s_logit[tid] = (maskp[b * Hc + tid] != 0) ? acc : NEGV;
  }
  __syncthreads();

  if (tid == 0) {
    float mx = s_logit[0];
    for (int h = 1; h < Hc; ++h) mx = fmaxf(mx, s_logit[h]);
    float sum = 0.0f;
    for (int h = 0; h < Hc; ++h) sum += __expf(s_logit[h] - mx);
    s_red[0] = mx;
    s_red[1] = 1.0f / sum;
  }
  __syncthreads();

  if (tid < Hc) s_alpha[tid] = __expf(s_logit[tid] - s_red[0]) * s_red[1];
  __syncthreads();

  const float* histb = hist + (size_t)b * Hc * Fc;
  float* outrow = out + (size_t)bn * Fc;
  for (int f = tid; f < Fc; f += 64) {
    float acc = 0.0f;
#pragma unroll
    for (int h = 0; h < Hc; ++h)
      acc = fmaf(s_alpha[h], histb[h * Fc + f], acc);
    outrow[f] = acc;
  }
}

extern "C" void kernel_launch(void* const* d_in, const int* in_sizes, int n_in,
                              void* d_out, int out_size, void* d_ws, size_t ws_size,
                              hipStream_t stream) {
  const float* history   = (const float*)d_in[0];  // (B,H,D)
  const float* cand_repr = (const float*)d_in[1];  // (B,N,D)
  const int*   maskp     = (const int*)d_in[2];    // (B,H) int32
  const int*   posp      = (const int*)d_in[3];    // (B,H) int (harness: int*)
  const float* pos_emb   = (const float*)d_in[4];  // (52,P)
  const float* W1        = (const float*)d_in[5];  // (A, 2F)
  const float* b1        = (const float*)d_in[6];  // (A,)
  const float* w2        = (const float*)d_in[7];  // (A,)
  const float* b2        = (const float*)d_in[8];  // (1,)

  float* out      = (float*)d_out;                 // user_repr (B*N*500)
  float* cand_out = out + (size_t)Bc * Nc * Fc;    // cand      (B*N*500)

  float* ws    = (float*)d_ws;
  float* hist  = ws;                               // B*H*500 = 1,600,000
  float* hcbuf = hist + (size_t)Bc * Hc * Fc;      // B*N*200 =   640,000
  float* hhbuf = hcbuf + (size_t)Bc * Nc * Ac;     // B*H*200 =   640,000

  const int totH = Bc * Hc * Fc;
  build_hist_kernel<<<(totH + 255) / 256, 256, 0, stream>>>(history, posp, pos_emb, hist);

  const int totC = Bc * Nc * Fc;
  build_cand_kernel<<<(totC + 255) / 256, 256, 0, stream>>>(cand_repr, pos_emb, cand_out);

  dim3 g((Bc * Nc) / 16, (Ac + 15) / 16, 2);       // (200, 13, 2)
  gemm_wmma_kernel<<<g, 32, 0, stream>>>(cand_out, hist, W1, hcbuf, hhbuf);

  attn_kernel<<<Bc * Nc, 64, 0, stream>>>(hcbuf, hhbuf, hist, maskp, b1, w2, b2, out);
}